// MHNF_87514253623562
// MI455X (gfx1250) — compile-verified
//
#include <hip/hip_runtime.h>
#include <hip/hip_bf16.h>

typedef float v2f __attribute__((ext_vector_type(2)));
typedef float v8f __attribute__((ext_vector_type(8)));

namespace {
constexpr int   Nn_   = 2048;
constexpr int   Rr    = 4;
constexpr int   Pp    = 2;
constexpr int   Ff    = 256;
constexpr int   Hd    = 256;
constexpr int   Oo    = 16;
constexpr int   NCc   = 1024;
constexpr long long NN = (long long)Nn_ * Nn_;
constexpr float EPSv  = 1e-6f;

constexpr int TM = 128, TN = 128, TK = 16;
constexpr int LSA = TK + 2;   // LDS A row stride (floats), keeps b64 alignment (even)
constexpr int LSB = TK + 2;   // LDS B^T row stride (floats), even -> aligned v2f reads

enum { OP_NONE = 0, OP_MASK = 1, OP_RELU_BIAS = 2 };
}

// ---------------------------------------------------------------------------
// Generic batched GEMM, fp32 via V_WMMA_F32_16X16X4_F32.
// C[b] = A[(b & aMask)] (MxK, row major) @ B[(b & bMask)] (KxN, row major)
// epilogue: OP_MASK -> (x>0?1:0), OP_RELU_BIAS -> relu(x + bias[(b&biasMask)*N + col])
// Requires: M % 128 == 0, N % 128 == 0, K % 16 == 0 (true for all uses here).
// Block tile 128x128, 8 waves (4x2), wave tile 32x64 (2x4 accumulators).
// A panel stored row-major [128][16]; B panel stored TRANSPOSED [128][16] so both
// WMMA fragments are single contiguous b64 LDS reads (no repacking movs).
// ---------------------------------------------------------------------------
__global__ __launch_bounds__(256)
void gemm_f32_wmma(const float* __restrict__ Abase, const float* __restrict__ Bbase,
                   float* __restrict__ Cbase, int M, int K, int Nc,
                   long long strideA, long long strideB, long long strideC,
                   int aMask, int bMask, int epilogue,
                   const float* __restrict__ bias, int biasMask)
{
    __shared__ float lA [TM * LSA];   // A  panel: [row][k]
    __shared__ float lBT[TN * LSB];   // B^T panel: [col][k]

    const int b = blockIdx.z;
    const float* A = Abase + (long long)(b & aMask) * strideA;
    const float* B = Bbase + (long long)(b & bMask) * strideB;
    float* C = Cbase + (long long)b * strideC;

    const int mBase = blockIdx.y * TM;
    const int nBase = blockIdx.x * TN;
    const int tid   = threadIdx.x;
    const int lane  = tid & 31;
    const int wid   = tid >> 5;          // 0..7
    const int m0    = (wid & 3) * 32;    // wave row offset in block tile
    const int n0    = (wid >> 2) * 64;   // wave col offset in block tile
    const int lh    = lane >> 4;         // lane half (K select)
    const int l15   = lane & 15;

    v8f acc[2][4] = {};

    for (int k0 = 0; k0 < K; k0 += TK) {
        // stage A panel: 128 rows x 16 cols = 512 float4, 2 per thread
        #pragma unroll
        for (int it = 0; it < 2; ++it) {
            int idx = tid + it * 256;
            int row = idx >> 2;
            int c4  = (idx & 3) << 2;
            float4 v = *(const float4*)(A + (long long)(mBase + row) * K + k0 + c4);
            float2* d = (float2*)&lA[row * LSA + c4];
            d[0] = make_float2(v.x, v.y);
            d[1] = make_float2(v.z, v.w);
        }
        // stage B panel transposed: 16 rows x 128 cols = 512 float4, 2 per thread
        #pragma unroll
        for (int it = 0; it < 2; ++it) {
            int idx = tid + it * 256;
            int row = idx >> 5;            // k index 0..15
            int c4  = (idx & 31) << 2;     // col 0..124
            float4 v = *(const float4*)(B + (long long)(k0 + row) * Nc + nBase + c4);
            lBT[(c4 + 0) * LSB + row] = v.x;
            lBT[(c4 + 1) * LSB + row] = v.y;
            lBT[(c4 + 2) * LSB + row] = v.z;
            lBT[(c4 + 3) * LSB + row] = v.w;
        }
        // prefetch next K panel into cache while this one is consumed
        if (k0 + TK < K) {
            __builtin_prefetch(A + (long long)(mBase + (tid >> 1)) * K + (k0 + TK), 0, 0);
            __builtin_prefetch(B + (long long)(k0 + TK + (tid >> 4)) * Nc + nBase + ((tid & 15) << 3), 0, 0);
        }
        __syncthreads();

        #pragma unroll
        for (int kk = 0; kk < TK; kk += 4) {
            const int ka = kk + 2 * lh;   // K index for this lane half
            v2f a0 = *(const v2f*)&lA[(m0 + l15) * LSA + ka];
            v2f a1 = *(const v2f*)&lA[(m0 + 16 + l15) * LSA + ka];
            v2f b0 = *(const v2f*)&lBT[(n0 + l15) * LSB + ka];
            v2f b1 = *(const v2f*)&lBT[(n0 + 16 + l15) * LSB + ka];
            v2f b2 = *(const v2f*)&lBT[(n0 + 32 + l15) * LSB + ka];
            v2f b3 = *(const v2f*)&lBT[(n0 + 48 + l15) * LSB + ka];

            acc[0][0] = __builtin_amdgcn_wmma_f32_16x16x4_f32(false, a0, false, b0, (short)0, acc[0][0], false, false);
            acc[0][1] = __builtin_amdgcn_wmma_f32_16x16x4_f32(false, a0, false, b1, (short)0, acc[0][1], false, false);
            acc[0][2] = __builtin_amdgcn_wmma_f32_16x16x4_f32(false, a0, false, b2, (short)0, acc[0][2], false, false);
            acc[0][3] = __builtin_amdgcn_wmma_f32_16x16x4_f32(false, a0, false, b3, (short)0, acc[0][3], false, false);
            acc[1][0] = __builtin_amdgcn_wmma_f32_16x16x4_f32(false, a1, false, b0, (short)0, acc[1][0], false, false);
            acc[1][1] = __builtin_amdgcn_wmma_f32_16x16x4_f32(false, a1, false, b1, (short)0, acc[1][1], false, false);
            acc[1][2] = __builtin_amdgcn_wmma_f32_16x16x4_f32(false, a1, false, b2, (short)0, acc[1][2], false, false);
            acc[1][3] = __builtin_amdgcn_wmma_f32_16x16x4_f32(false, a1, false, b3, (short)0, acc[1][3], false, false);
        }
        __syncthreads();
    }

    // epilogue + store; C/D layout: row = base + 8*laneHalf + vgpr, col = base + lane15
    const float* bv = (epilogue == OP_RELU_BIAS) ? bias + (long long)(b & biasMask) * Nc : nullptr;
    #pragma unroll
    for (int mi = 0; mi < 2; ++mi) {
        #pragma unroll
        for (int ni = 0; ni < 4; ++ni) {
            int r0 = mBase + m0 + mi * 16 + lh * 8;
            int c  = nBase + n0 + ni * 16 + l15;
            v8f v = acc[mi][ni];
            #pragma unroll
            for (int g = 0; g < 8; ++g) {
                float x = v[g];
                if (epilogue == OP_MASK)           x = (x > 0.0f) ? 1.0f : 0.0f;
                else if (epilogue == OP_RELU_BIAS) { x += bv[c]; x = x > 0.0f ? x : 0.0f; }
                C[(long long)(r0 + g) * Nc + c] = x;
            }
        }
    }
}

// ---------------------------------------------------------------------------
// GTConv for W1/W2/Wl + union mask + fused exp-on-edges for the sm_norm inputs.
// ---------------------------------------------------------------------------
__global__ __launch_bounds__(256)
void gtconv_kernel(const float* __restrict__ adj, const unsigned char* __restrict__ mask,
                   const float* __restrict__ W1, const float* __restrict__ W2,
                   const float* __restrict__ Wl,
                   float* __restrict__ wG, float* __restrict__ wB,
                   float* __restrict__ BL, float* __restrict__ umf)
{
    long long i = (long long)blockIdx.x * blockDim.x + threadIdx.x;
    if (i >= NN) return;
    float s1[Pp] = {0.f, 0.f}, s2[Pp] = {0.f, 0.f}, sl[Pp] = {0.f, 0.f};
    bool un = false;
    #pragma unroll
    for (int r = 0; r < Rr; ++r) {
        bool m = mask[r * NN + i] != 0;
        if (m) {
            un = true;
            float a = adj[r * NN + i];
            #pragma unroll
            for (int p = 0; p < Pp; ++p) {
                s1[p] += W1[p * Rr + r] * a;
                s2[p] += W2[p * Rr + r] * a;
                sl[p] += Wl[p * Rr + r] * a;
            }
        }
    }
    umf[i] = un ? 1.0f : 0.0f;
    #pragma unroll
    for (int p = 0; p < Pp; ++p) {
        wG[p * NN + i] = un ? __expf(s1[p]) : 0.0f;
        wB[p * NN + i] = un ? __expf(s2[p]) : 0.0f;
        BL[p * NN + i] = sl[p];
    }
}

// column sums over u (axis 2 of [B,N,N]) -> rcol[b,v] = 1/(sum+eps)
__global__ __launch_bounds__(256)
void colsum_kernel(const float* __restrict__ in, float* __restrict__ rcol)
{
    int v = blockIdx.x * blockDim.x + threadIdx.x;
    int pb = blockIdx.y;
    const float* base = in + (long long)pb * NN;
    float s = 0.f;
    for (int u = 0; u < Nn_; ++u) s += base[(long long)u * Nn_ + v];
    rcol[(long long)pb * Nn_ + v] = 1.0f / (s + EPSv);
}

// in[b,u,v] *= rcol[b,v]
__global__ __launch_bounds__(256)
void scalecol_kernel(float* __restrict__ in, const float* __restrict__ rcol, long long total)
{
    long long i = (long long)blockIdx.x * blockDim.x + threadIdx.x;
    if (i >= total) return;
    long long pb = i / NN;
    int v = (int)(i % Nn_);
    in[i] *= rcol[pb * Nn_ + v];
}

// G[j,p,u,v] *= (M_j[u,v] > 0 && u != v); G contiguous [3,2,N,N], Mbuf [3,N,N]
__global__ __launch_bounds__(256)
void maskdiag_kernel(float* __restrict__ G, const float* __restrict__ Mbuf)
{
    long long i = (long long)blockIdx.x * blockDim.x + threadIdx.x;
    if (i >= 6 * NN) return;
    long long s = i / NN;          // j*2+p
    long long mi = i % NN;
    int u = (int)(mi / Nn_), v = (int)(mi % Nn_);
    const float* m = Mbuf + (s >> 1) * NN;
    if (!(m[mi] > 0.0f) || u == v) G[i] = 0.0f;
}

// alpha[jp,u] = relu(att_v[p] * tanh(dot_h(H[jp,u,:], att_w[p,:]) + att_b[p]))
__global__ __launch_bounds__(256)
void att_layer_kernel(const float* __restrict__ H, const float* __restrict__ att_w,
                      const float* __restrict__ att_b, const float* __restrict__ att_v,
                      float* __restrict__ alpha)
{
    int gw = (blockIdx.x * blockDim.x + threadIdx.x) >> 5;   // one wave per (jp,u)
    int lane = threadIdx.x & 31;
    int u = gw % Nn_, jp = gw / Nn_, p = jp % Pp;
    const float* h = H + ((long long)jp * Nn_ + u) * Hd;
    const float* w = att_w + p * Hd;
    float s = 0.f;
    for (int i = lane; i < Hd; i += 32) s += h[i] * w[i];
    for (int off = 16; off; off >>= 1) s += __shfl_xor(s, off, 32);
    if (lane == 0) {
        float t = att_v[p] * tanhf(s + att_b[p]);
        alpha[gw] = t > 0.f ? t : 0.f;
    }
}

// beta = softmax over j (J=3) of alpha[j*P+p, u]
__global__ __launch_bounds__(256)
void beta_kernel(const float* __restrict__ alpha, float* __restrict__ beta)
{
    int i = blockIdx.x * blockDim.x + threadIdx.x;   // (p,u) flat, size P*N
    if (i >= Pp * Nn_) return;
    int p = i / Nn_, u = i % Nn_;
    float a0 = alpha[(0 * Pp + p) * Nn_ + u];
    float a1 = alpha[(1 * Pp + p) * Nn_ + u];
    float a2 = alpha[(2 * Pp + p) * Nn_ + u];
    float m = fmaxf(a0, fmaxf(a1, a2));
    float e0 = __expf(a0 - m), e1 = __expf(a1 - m), e2 = __expf(a2 - m);
    float r = 1.0f / (e0 + e1 + e2);
    beta[(0 * Pp + p) * Nn_ + u] = e0 * r;
    beta[(1 * Pp + p) * Nn_ + u] = e1 * r;
    beta[(2 * Pp + p) * Nn_ + u] = e2 * r;
}

// mp_feat[p,u,h] = sum_j H[j,p,u,h] * beta[j,p,u]
__global__ __launch_bounds__(256)
void mpfeat_kernel(const float* __restrict__ H, const float* __restrict__ beta,
                   float* __restrict__ mpf)
{
    long long i = (long long)blockIdx.x * blockDim.x + threadIdx.x;  // (p,u,h)
    if (i >= (long long)Pp * Nn_ * Hd) return;
    long long pu = i / Hd;     // p*N+u
    int h = (int)(i % Hd);
    long long p = pu / Nn_, u = pu % Nn_;
    float s = 0.f;
    #pragma unroll
    for (int j = 0; j < 3; ++j) {
        long long jp = (long long)j * Pp + p;
        s += H[(jp * Nn_ + u) * Hd + h] * beta[jp * Nn_ + u];
    }
    mpf[i] = s;
}

// a[p,u] = relu(mp_v * tanh(dot_h(mp_feat[p,u,:], mp_w) + mp_b))
__global__ __launch_bounds__(256)
void att_mp_kernel(const float* __restrict__ mpf, const float* __restrict__ mp_w,
                   const float* __restrict__ mp_b, const float* __restrict__ mp_v,
                   float* __restrict__ avec)
{
    int gw = (blockIdx.x * blockDim.x + threadIdx.x) >> 5;   // one wave per (p,u)
    int lane = threadIdx.x & 31;
    const float* f = mpf + (long long)gw * Hd;
    float s = 0.f;
    for (int i = lane; i < Hd; i += 32) s += f[i] * mp_w[i];
    for (int off = 16; off; off >>= 1) s += __shfl_xor(s, off, 32);
    if (lane == 0) {
        float t = mp_v[0] * tanhf(s + mp_b[0]);
        avec[gw] = t > 0.f ? t : 0.f;
    }
}

// fused[u,h] = sum_p mp_feat[p,u,h] * softmax_p(a[p,u])
__global__ __launch_bounds__(256)
void fuse_kernel(const float* __restrict__ mpf, const float* __restrict__ avec,
                 float* __restrict__ fused)
{
    long long i = (long long)blockIdx.x * blockDim.x + threadIdx.x;  // (u,h)
    if (i >= (long long)Nn_ * Hd) return;
    long long u = i / Hd;
    float a0 = avec[u], a1 = avec[Nn_ + u];
    float m = fmaxf(a0, a1);
    float e0 = __expf(a0 - m), e1 = __expf(a1 - m);
    float r = 1.0f / (e0 + e1);
    fused[i] = mpf[i] * (e0 * r) + mpf[(long long)Nn_ * Hd + i] * (e1 * r);
}

// out[c,o] = sum_h fused[nids[c],h] * lin_w[h,o] + lin_b[o]
__global__ __launch_bounds__(256)
void final_kernel(const float* __restrict__ fused, const float* __restrict__ lin_w,
                  const float* __restrict__ lin_b, const int* __restrict__ nids,
                  float* __restrict__ out)
{
    int i = blockIdx.x * blockDim.x + threadIdx.x;   // (c,o)
    if (i >= NCc * Oo) return;
    int c = i / Oo, o = i % Oo;
    const float* f = fused + (long long)nids[c] * Hd;
    float s = lin_b[o];
    for (int h = 0; h < Hd; ++h) s += f[h] * lin_w[h * Oo + o];
    out[i] = s;
}

// ---------------------------------------------------------------------------
extern "C" void kernel_launch(void* const* d_in, const int* in_sizes, int n_in,
                              void* d_out, int out_size, void* d_ws, size_t ws_size,
                              hipStream_t stream)
{
    const float*         adj   = (const float*)d_in[0];
    const unsigned char* mask  = (const unsigned char*)d_in[1];
    const float*         feat  = (const float*)d_in[2];
    const float*         W1    = (const float*)d_in[3];
    const float*         W2    = (const float*)d_in[4];
    const float*         Wl    = (const float*)d_in[5];
    const float*         gcn_w = (const float*)d_in[6];
    const float*         gcn_b = (const float*)d_in[7];
    const float*         att_w = (const float*)d_in[8];
    const float*         att_b = (const float*)d_in[9];
    const float*         att_v = (const float*)d_in[10];
    const float*         mp_w  = (const float*)d_in[11];
    const float*         mp_b  = (const float*)d_in[12];
    const float*         mp_v  = (const float*)d_in[13];
    const float*         lin_w = (const float*)d_in[14];
    const float*         lin_b = (const float*)d_in[15];
    const int*           nids  = (const int*)d_in[16];
    float* out = (float*)d_out;

    // workspace layout (floats)
    float* G     = (float*)d_ws;                    // [3,2,N,N]: G0/Wn0, G1/Wn1, G2/Wn2
    float* RB    = G    + 6 * NN;                   // [2,N,N]
    float* BL    = RB   + 2 * NN;                   // [2,N,N]
    float* Mbuf  = BL   + 2 * NN;                   // [3,N,N]: umf, M1f, M2f
    float* rcol  = Mbuf + 3 * NN;                   // [6,N]
    float* hW    = rcol + 6LL * Nn_;                // [2,N,Hd]
    float* Hb    = hW   + (long long)Pp * Nn_ * Hd; // [3,2,N,Hd]
    float* alpha = Hb   + 6LL * Nn_ * Hd;           // [6,N]
    float* beta  = alpha + 6LL * Nn_;               // [6,N]
    float* avec  = beta  + 6LL * Nn_;               // [2,N]
    float* mpf   = avec  + 2LL * Nn_;               // [2,N,Hd]
    float* fused = mpf   + (long long)Pp * Nn_ * Hd;// [N,Hd]

    const int B256 = 256;
    const long long PNN = 2 * NN;

    // 1) gtconv + exp-on-edges + union mask
    gtconv_kernel<<<dim3((unsigned)((NN + 255) / 256)), B256, 0, stream>>>(
        adj, mask, W1, W2, Wl, G, RB, BL, Mbuf);

    // 2) sm_norm for RA (in G[0..2NN)) and RB
    colsum_kernel<<<dim3(Nn_ / 256, 2), B256, 0, stream>>>(G, rcol);
    scalecol_kernel<<<dim3((unsigned)(PNN / 256)), B256, 0, stream>>>(G, rcol, PNN);
    colsum_kernel<<<dim3(Nn_ / 256, 2), B256, 0, stream>>>(RB, rcol);
    scalecol_kernel<<<dim3((unsigned)(PNN / 256)), B256, 0, stream>>>(RB, rcol, PNN);

    // 3) G1 = RA @ RB, G2 = G1 @ BL  (batched over P; masks: b&1 selects p)
    gemm_f32_wmma<<<dim3(Nn_ / TN, Nn_ / TM, 2), B256, 0, stream>>>(
        G, RB, G + 2 * NN, Nn_, Nn_, Nn_, NN, NN, NN, 1, 1, OP_NONE, nullptr, 0);
    gemm_f32_wmma<<<dim3(Nn_ / TN, Nn_ / TM, 2), B256, 0, stream>>>(
        G + 2 * NN, BL, G + 4 * NN, Nn_, Nn_, Nn_, NN, NN, NN, 1, 1, OP_NONE, nullptr, 0);

    // 4) mask propagation: M1 = (umf@umf)>0, M2 = (M1f@umf)>0
    gemm_f32_wmma<<<dim3(Nn_ / TN, Nn_ / TM, 1), B256, 0, stream>>>(
        Mbuf, Mbuf, Mbuf + NN, Nn_, Nn_, Nn_, NN, NN, NN, 0, 0, OP_MASK, nullptr, 0);
    gemm_f32_wmma<<<dim3(Nn_ / TN, Nn_ / TM, 1), B256, 0, stream>>>(
        Mbuf + NN, Mbuf, Mbuf + 2 * NN, Nn_, Nn_, Nn_, NN, NN, NN, 0, 0, OP_MASK, nullptr, 0);

    // 5) Wd = G * (M & ~eye), then right-norm over u -> Wn (in place)
    maskdiag_kernel<<<dim3((unsigned)(6 * NN / 256)), B256, 0, stream>>>(G, Mbuf);
    colsum_kernel<<<dim3(Nn_ / 256, 6), B256, 0, stream>>>(G, rcol);
    scalecol_kernel<<<dim3((unsigned)(6 * NN / 256)), B256, 0, stream>>>(G, rcol, 6 * NN);

    // 6) hW[p] = feat @ gcn_w[p]  (A shared: mask 0, stride 0)
    gemm_f32_wmma<<<dim3(Hd / TN, Nn_ / TM, 2), B256, 0, stream>>>(
        feat, gcn_w, hW, Nn_, Ff, Hd, 0, (long long)Ff * Hd, (long long)Nn_ * Hd,
        0, 1, OP_NONE, nullptr, 0);

    // 7) H[j,p] = relu(Wn[j,p] @ hW[p] + gcn_b[p])  (batch 6; b&7==b, B/bias via b&1)
    gemm_f32_wmma<<<dim3(Hd / TN, Nn_ / TM, 6), B256, 0, stream>>>(
        G, hW, Hb, Nn_, Nn_, Hd, NN, (long long)Nn_ * Hd, (long long)Nn_ * Hd,
        7, 1, OP_RELU_BIAS, gcn_b, 1);

    // 8) layer attention -> beta, mp_feat
    att_layer_kernel<<<dim3(6 * Nn_ * 32 / 256), B256, 0, stream>>>(Hb, att_w, att_b, att_v, alpha);
    beta_kernel<<<dim3(Pp * Nn_ / 256), B256, 0, stream>>>(alpha, beta);
    mpfeat_kernel<<<dim3((unsigned)((long long)Pp * Nn_ * Hd / 256)), B256, 0, stream>>>(Hb, beta, mpf);

    // 9) meta-path attention -> fused
    att_mp_kernel<<<dim3(Pp * Nn_ * 32 / 256), B256, 0, stream>>>(mpf, mp_w, mp_b, mp_v, avec);
    fuse_kernel<<<dim3((unsigned)((long long)Nn_ * Hd / 256)), B256, 0, stream>>>(mpf, avec, fused);

    // 10) final linear + gather
    final_kernel<<<dim3(NCc * Oo / 256), B256, 0, stream>>>(fused, lin_w, lin_b, nids, out);
}